// SAGE_652835029772
// MI455X (gfx1250) — compile-verified
//
#include <hip/hip_runtime.h>

typedef float v2f __attribute__((ext_vector_type(2)));
typedef float v8f __attribute__((ext_vector_type(8)));

#define IN_F  64
#define HID_F 64
#define OUT_F 32

// ---------------------------------------------------------------- fill zeros
__global__ __launch_bounds__(256) void fill_zero_f32(float* __restrict__ p, int n) {
  int i = blockIdx.x * blockDim.x + threadIdx.x;
  if (i < n) p[i] = 0.0f;
}

// ---------------------------------------------------------------- in-degree
__global__ __launch_bounds__(256) void degree_kernel(const int* __restrict__ dst,
                                                     float* __restrict__ deg, int nE) {
  int e = blockIdx.x * blockDim.x + threadIdx.x;
  if (e < nE)
    __hip_atomic_fetch_add(&deg[dst[e]], 1.0f, __ATOMIC_RELAXED, __HIP_MEMORY_SCOPE_AGENT);
}

// ---------------------------------------------------------------- 1/max(deg,1)
__global__ __launch_bounds__(256) void recip_kernel(const float* __restrict__ deg,
                                                    float* __restrict__ inv, int n) {
  int i = blockIdx.x * blockDim.x + threadIdx.x;
  if (i < n) inv[i] = 1.0f / fmaxf(deg[i], 1.0f);
}

// ---------------------------------------------------------------- edge scatter-add
// 16 threads per edge, 4 features per thread (float4 gather, 4 fp32 atomics).
// Feature matrices (25.6 MB) live in the 192 MB L2, so the random gather and
// the atomic adds both resolve in L2.
__global__ __launch_bounds__(256) void scatter_add64(const float* __restrict__ feat,
                                                     const int* __restrict__ src,
                                                     const int* __restrict__ dst,
                                                     float* __restrict__ agg, int nE) {
  int tid = blockIdx.x * blockDim.x + threadIdx.x;
  int e = tid >> 4;
  if (e >= nE) return;
  int f = (tid & 15) << 2;
  int s = src[e];
  int d = dst[e];
  const float4 v = *(const float4*)(feat + (size_t)s * 64 + f);
  float* p = agg + (size_t)d * 64 + f;
  __hip_atomic_fetch_add(p + 0, v.x, __ATOMIC_RELAXED, __HIP_MEMORY_SCOPE_AGENT);
  __hip_atomic_fetch_add(p + 1, v.y, __ATOMIC_RELAXED, __HIP_MEMORY_SCOPE_AGENT);
  __hip_atomic_fetch_add(p + 2, v.z, __ATOMIC_RELAXED, __HIP_MEMORY_SCOPE_AGENT);
  __hip_atomic_fetch_add(p + 3, v.w, __ATOMIC_RELAXED, __HIP_MEMORY_SCOPE_AGENT);
}

// ---------------------------------------------------------------- fused SAGE GEMM
// Hout[16-row tile] = act( Hin @ Wself + (Agg * inv) @ Wneigh + bias )
// One wave32 per 16-node tile. K = 64, NOUT = 64 (layer1) or 32 (layer2).
// Uses V_WMMA_F32_16X16X4_F32 (full fp32 matrix core path).
template <int NOUT, bool RELU>
__global__ __launch_bounds__(256) void sage_gemm(const float* __restrict__ Hin,   // [N,64]
                                                 const float* __restrict__ Agg,   // [N,64]
                                                 const float* __restrict__ inv,   // [N]
                                                 const float* __restrict__ Wself, // [64,NOUT]
                                                 const float* __restrict__ Wneigh,// [64,NOUT]
                                                 const float* __restrict__ bias,  // [NOUT]
                                                 float* __restrict__ Hout,        // [N,NOUT]
                                                 int nTiles) {
  constexpr int K  = 64;
  constexpr int NT = NOUT / 16;

  __builtin_prefetch(Wself, 0, 1);
  __builtin_prefetch(Wneigh, 0, 1);

  int wave = (blockIdx.x * blockDim.x + threadIdx.x) >> 5;
  if (wave >= nTiles) return;                 // uniform per wave: EXEC stays all-1s
  int lane = threadIdx.x & 31;
  int hi   = lane >> 4;                       // 0: lanes 0-15, 1: lanes 16-31
  int lo   = lane & 15;
  int row  = wave * 16 + lo;                  // node row this lane carries (A layout)
  float invd = inv[row];

  v8f acc[NT] = {};

#pragma unroll
  for (int k = 0; k < K; k += 4) {
    // --- A fragments (16x4 fp32): vgpr0 = K{k+2*hi}, vgpr1 = K{k+2*hi+1}
    float2 fs = *(const float2*)(Hin + (size_t)row * K + k + 2 * hi);
    v2f aS; aS.x = fs.x; aS.y = fs.y;
    float2 fn = *(const float2*)(Agg + (size_t)row * K + k + 2 * hi);
    v2f aN; aN.x = fn.x * invd; aN.y = fn.y * invd;   // mean aggregation folded in

#pragma unroll
    for (int t = 0; t < NT; ++t) {
      int col = t * 16 + lo;
      // --- B fragments (4x16): vgpr0 = row k+2*hi, vgpr1 = row k+2*hi+1
      const float* wS = Wself  + (size_t)(k + 2 * hi) * NOUT + col;
      v2f bS; bS.x = wS[0]; bS.y = wS[NOUT];
      const float* wN = Wneigh + (size_t)(k + 2 * hi) * NOUT + col;
      v2f bN; bN.x = wN[0]; bN.y = wN[NOUT];

      acc[t] = __builtin_amdgcn_wmma_f32_16x16x4_f32(false, aS, false, bS,
                                                     (short)0, acc[t], false, false);
      acc[t] = __builtin_amdgcn_wmma_f32_16x16x4_f32(false, aN, false, bN,
                                                     (short)0, acc[t], false, false);
    }
  }

  // --- store: C/D layout, vgpr r holds rows {r, r+8} with cols striped on lanes
#pragma unroll
  for (int t = 0; t < NT; ++t) {
    int col = t * 16 + lo;
    float b = bias[col];
#pragma unroll
    for (int r = 0; r < 8; ++r) {
      int m = r + 8 * hi;
      float v = acc[t][r] + b;
      if (RELU) v = fmaxf(v, 0.0f);
      Hout[(size_t)(wave * 16 + m) * NOUT + col] = v;
    }
  }
}

// ---------------------------------------------------------------- launch
extern "C" void kernel_launch(void* const* d_in, const int* in_sizes, int n_in,
                              void* d_out, int out_size, void* d_ws, size_t ws_size,
                              hipStream_t stream) {
  const float* x       = (const float*)d_in[0];
  const int*   src     = (const int*)  d_in[1];
  const int*   dst     = (const int*)  d_in[2];
  const float* Wself1  = (const float*)d_in[3];
  const float* Wneigh1 = (const float*)d_in[4];
  const float* b1      = (const float*)d_in[5];
  const float* Wself2  = (const float*)d_in[6];
  const float* Wneigh2 = (const float*)d_in[7];
  const float* b2      = (const float*)d_in[8];
  float* out = (float*)d_out;

  const int N = in_sizes[0] / IN_F;   // 100000 (multiple of 16)
  const int E = in_sizes[1];          // 1600000

  // Workspace layout (floats): deg[N] | agg[64N] | h[64N] | inv[N]  (~52 MB)
  float* ws  = (float*)d_ws;
  float* deg = ws;
  float* agg = ws + N;
  float* h   = ws + N + (size_t)N * 64;
  float* inv = ws + N + (size_t)N * 128;

  const int TPB = 256;
  int nTiles = N / 16;
  int gemmBlocks = (nTiles + 7) / 8;          // 8 waves per 256-thread block
  int scatBlocks = (E * 16 + TPB - 1) / TPB;  // 16 threads per edge

  // 1) zero deg + agg (contiguous 65N floats)
  {
    int n = 65 * N;
    fill_zero_f32<<<(n + TPB - 1) / TPB, TPB, 0, stream>>>(deg, n);
  }
  // 2) in-degree, 3) reciprocal of clamped degree
  degree_kernel<<<(E + TPB - 1) / TPB, TPB, 0, stream>>>(dst, deg, E);
  recip_kernel<<<(N + TPB - 1) / TPB, TPB, 0, stream>>>(deg, inv, N);

  // 4) layer-1 neighbor sum, 5) fused GEMM + bias + ReLU
  scatter_add64<<<scatBlocks, TPB, 0, stream>>>(x, src, dst, agg, E);
  sage_gemm<HID_F, true><<<gemmBlocks, TPB, 0, stream>>>(x, agg, inv, Wself1, Wneigh1,
                                                         b1, h, nTiles);

  // 6) re-zero agg, 7) layer-2 neighbor sum, 8) fused GEMM + bias
  {
    int n = 64 * N;
    fill_zero_f32<<<(n + TPB - 1) / TPB, TPB, 0, stream>>>(agg, n);
  }
  scatter_add64<<<scatBlocks, TPB, 0, stream>>>(h, src, dst, agg, E);
  sage_gemm<OUT_F, false><<<gemmBlocks, TPB, 0, stream>>>(h, agg, inv, Wself2, Wneigh2,
                                                          b2, out, nTiles);
}